// ProbabilisticSkipper_7773890806432
// MI455X (gfx1250) — compile-verified
//
#include <hip/hip_runtime.h>
#include <hip/hip_bf16.h>
#include <stdint.h>

#define BATCH 32
#define SEQ   4096
#define DIM   256   // floats per timestep = 1 KiB

// ---------------------------------------------------------------------------
// Kernel A: per-row exclusive prefix sum of keep_mask.
// One 256-thread block per batch row; each thread owns 16 consecutive mask
// entries. Block scan done in LDS (Hillis-Steele over 256 partials).
// Outputs: scan[b*SEQ + t] = #kept strictly before t   (exclusive)
//          counts[b]       = total kept in row b
// ---------------------------------------------------------------------------
__global__ void scan_rows(const int* __restrict__ keep,
                          int* __restrict__ scan,
                          int* __restrict__ counts) {
    __shared__ int smem[256];
    const int b    = blockIdx.x;
    const int tid  = threadIdx.x;
    const int base = b * SEQ + tid * 16;

    int v[16];
    int sum = 0;
#pragma unroll
    for (int i = 0; i < 16; ++i) {
        v[i] = keep[base + i];
        sum += v[i];
    }

    smem[tid] = sum;
    __syncthreads();
#pragma unroll
    for (int ofs = 1; ofs < 256; ofs <<= 1) {
        int t = (tid >= ofs) ? smem[tid - ofs] : 0;
        __syncthreads();
        smem[tid] += t;
        __syncthreads();
    }

    int run = smem[tid] - sum;   // exclusive prefix for this thread's chunk
#pragma unroll
    for (int i = 0; i < 16; ++i) {
        scan[base + i] = run;
        run += v[i];
    }
    if (tid == 255) counts[b] = smem[255];
}

// ---------------------------------------------------------------------------
// Kernel B: one 64-thread block (2 wave32s) per (b, t) source slot.
//   kept   : copy x[b,t,:] (1 KiB) -> out[b, scan[b,t], :] via the CDNA5
//            async copy path: global_load_async_to_lds_b128 (16 B/lane),
//            s_wait_asynccnt 0, global_store_async_from_lds_b128.
//   dropped: this is the e-th dropped timestep (e = t - scan[b,t]); zero the
//            tail slot out[b, counts[b] + e, :]. Bijective -> every output
//            row written exactly once, no atomics, fully coalesced b128 ops.
// ---------------------------------------------------------------------------
__global__ void compact_copy(const float* __restrict__ x,
                             const int* __restrict__ keep,
                             const int* __restrict__ scan,
                             const int* __restrict__ counts,
                             float* __restrict__ out) {
    __shared__ float4 buf[64];           // 1 KiB staging, 16 B per lane

    const int slot = blockIdx.x;         // 0 .. BATCH*SEQ-1
    const int b    = slot >> 12;         // SEQ == 4096
    const int t    = slot & (SEQ - 1);
    const int lane = threadIdx.x;        // 0 .. 63
    const int idx  = b * SEQ + t;

    const int k  = keep[idx];
    const int kb = scan[idx];            // kept strictly before t

    if (k) {
        const int dst = kb;
        const uint64_t src = (uint64_t)(uintptr_t)(x   + (size_t)idx * DIM);
        const uint64_t dp  = (uint64_t)(uintptr_t)(out + ((size_t)b * SEQ + dst) * DIM);
        const unsigned voff = (unsigned)(lane * 16);                 // byte offset per lane
        const unsigned lds  = ((unsigned)(uintptr_t)buf) + voff;     // 32-bit LDS byte address

        // Memory -> LDS, 128 bits per lane (ASYNCcnt++)
        asm volatile("global_load_async_to_lds_b128 %0, %1, %2"
                     :: "v"(lds), "v"(voff), "s"(src) : "memory");
        // LDS must be populated before draining it
        asm volatile("s_wait_asynccnt 0x0" ::: "memory");
        // LDS -> Memory, 128 bits per lane (ASYNCcnt++); S_ENDPGM waits idle
        asm volatile("global_store_async_from_lds_b128 %0, %1, %2"
                     :: "v"(voff), "v"(lds), "s"(dp) : "memory");
    } else {
        const int dst = counts[b] + (t - kb);   // tail slot for e-th dropped row
        float4* o = (float4*)(out + ((size_t)b * SEQ + dst) * DIM);
        float4 z;
        z.x = 0.0f; z.y = 0.0f; z.z = 0.0f; z.w = 0.0f;
        o[lane] = z;
    }
}

// ---------------------------------------------------------------------------
// Launch: scan (32 blocks) then scatter/zero (131072 blocks x 64 threads).
// Scratch layout in d_ws: int scan[BATCH*SEQ]; int counts[BATCH];  (~513 KiB)
// ---------------------------------------------------------------------------
extern "C" void kernel_launch(void* const* d_in, const int* in_sizes, int n_in,
                              void* d_out, int out_size, void* d_ws, size_t ws_size,
                              hipStream_t stream) {
    const float* x    = (const float*)d_in[0];
    const int*   keep = (const int*)d_in[1];
    float*       out  = (float*)d_out;

    int* scan   = (int*)d_ws;
    int* counts = scan + BATCH * SEQ;

    scan_rows<<<BATCH, 256, 0, stream>>>(keep, scan, counts);
    compact_copy<<<BATCH * SEQ, 64, 0, stream>>>(x, keep, scan, counts, out);
}